// ChamferLoss_15977278341305
// MI455X (gfx1250) — compile-verified
//
#include <hip/hip_runtime.h>

typedef float v2f __attribute__((ext_vector_type(2)));
typedef float v8f __attribute__((ext_vector_type(8)));

#define BATCH 16
#define NPTS  4096
#define CST   6
#define NTILE (NPTS / 16)

// Monotonic order-preserving float <-> uint key (for unsigned atomicMin).
__device__ __forceinline__ unsigned encf(float f) {
    unsigned u = __float_as_uint(f);
    return u ^ (0x80000000u | (unsigned)((int)u >> 31));
}
__device__ __forceinline__ float decf(unsigned k) {
    unsigned mask = 0x80000000u | (unsigned)((~(int)k) >> 31);
    return __uint_as_float(k ^ mask);
}

__global__ void chamfer_init(unsigned* __restrict__ colKeys) {
    int idx = blockIdx.x * blockDim.x + threadIdx.x;
    if (idx < BATCH * NPTS) colKeys[idx] = 0xFFFFFFFFu;
}

// One workgroup = one batch b, 128 rows of x (8 wave32s x 16 rows).
// y staged once into LDS pre-swizzled into B-matrix lane layout:
//   tile jt, lane<16  -> (y0, y1)  = K0,K1 of column N=lane
//   tile jt, lane>=16 -> (y2, ry)  = K2,K3 of column N=lane-16
// Per 16-wide j tile (two K=4 WMMAs, C = inline 0):
//   D1 = (-2x | 1)  x (y | ry) = ry - 2 x.y -> row mins (rx added in reduce)
//   D2 = (-2x | rx) x (y | 1)  = rx - 2 x.y -> col mins (ry added in reduce)
__global__ __launch_bounds__(256) void chamfer_main(
    const float* __restrict__ x, const float* __restrict__ y,
    unsigned* __restrict__ colKeys, float* __restrict__ rowRaw) {

    __shared__ unsigned ldsCol[NPTS];              // 16 KB: col-min keys
    __shared__ float2   ldsB[(NTILE + 1) * 32];    // 64 KB + 1 pad tile

    const int tid     = threadIdx.x;
    const int lane    = tid & 31;
    const int wave    = tid >> 5;
    const int b       = blockIdx.x >> 5;           // 32 row-chunks per batch
    const int chunk   = blockIdx.x & 31;
    const int rowBase = chunk * 128 + wave * 16;

    const float* xb = x + (size_t)b * NPTS * CST;
    const float* yb = y + (size_t)b * NPTS * CST;

    // Stage y into swizzled B layout + init colmin keys.
    for (int j = tid; j < NPTS; j += 256) {
        ldsCol[j] = 0xFFFFFFFFu;
        float2 y01 = *(const float2*)(yb + j * CST);
        float  y2  = yb[j * CST + 2];
        float  ry  = fmaf(y2, y2, fmaf(y01.y, y01.y, y01.x * y01.x));
        const int t = j >> 4, c = j & 15;
        ldsB[t * 32 + c]      = make_float2(y01.x, y01.y);
        ldsB[t * 32 + 16 + c] = make_float2(y2, ry);
    }
    if (tid < 32) ldsB[NTILE * 32 + tid] = make_float2(0.0f, 0.0f); // pad
    __syncthreads();

    const int  mrow = lane & 15;
    const bool hi   = lane >= 16;                  // hi lanes: K2,K3 / M=8..15
    const int  row  = rowBase + mrow;

    float2 x01 = *(const float2*)(xb + row * CST);
    float  x2  = xb[row * CST + 2];
    float  rxl = fmaf(x2, x2, fmaf(x01.y, x01.y, x01.x * x01.x));

    // A layout (16x4 f32): v0 = K0(lo)/K2(hi), v1 = K1(lo)/K3(hi)
    const float ax    = -2.0f * (hi ? x2 : x01.x);
    const float ay_lo = -2.0f * x01.y;
    v2f a1, a2;
    a1.x = ax;  a1.y = hi ? 1.0f : ay_lo;          // K3 = 1
    a2.x = ax;  a2.y = hi ? rxl  : ay_lo;          // K3 = rx[M]

    v8f rowacc;
#pragma unroll
    for (int k = 0; k < 8; ++k) rowacc[k] = 3.4e38f;

    const float2* myB = ldsB + lane;               // per-lane swizzled stream
    float2 yv = myB[0];                            // prologue load

    for (int jt = 0; jt < NTILE; ++jt) {
        float2 yn = myB[(jt + 1) * 32];            // prefetch next tile (pad-safe)

        v2f b1, b2;
        b1.x = yv.x;  b1.y = yv.y;                 // straight from LDS layout
        b2.x = yv.x;  b2.y = hi ? 1.0f : yv.y;     // K3 = 1 (one cndmask)

        v8f cz = {};
        v8f d1 = __builtin_amdgcn_wmma_f32_16x16x4_f32(
            false, a1, false, b1, (short)0, cz, false, false);
        v8f d2 = __builtin_amdgcn_wmma_f32_16x16x4_f32(
            false, a2, false, b2, (short)0, cz, false, false);

        // Row direction: elementwise min of (ry - 2 x.y), stays in C-layout.
#pragma unroll
        for (int k = 0; k < 8; ++k) rowacc[k] = fminf(rowacc[k], d1[k]);

        // Col direction: min over this half's 8 rows of (rx - 2 x.y);
        // both lane halves ds_min the same address -> LDS atomic merges.
        float cmin = fminf(d2[0], d2[1]);
#pragma unroll
        for (int k = 2; k < 8; ++k) cmin = fminf(cmin, d2[k]);
        atomicMin(&ldsCol[jt * 16 + mrow], encf(cmin));

        yv = yn;
    }

    // Row mins: butterfly within each 16-lane half; store raw (ry-2xy) min.
#pragma unroll
    for (int k = 0; k < 8; ++k) {
        float r = rowacc[k];
#pragma unroll
        for (int d = 1; d < 16; d <<= 1) r = fminf(r, __shfl_xor(r, d, 32));
        if (mrow == 0) rowRaw[b * NPTS + rowBase + k + (hi ? 8 : 0)] = r;
    }

    __syncthreads();
    // Merge this chunk's column mins into global (min: order-independent).
    unsigned* gcol = colKeys + b * NPTS;
    for (int j = tid; j < NPTS; j += 256) atomicMin(&gcol[j], ldsCol[j]);
}

// Final fixed-order reduction; adds back the deferred rx / ry terms.
__global__ __launch_bounds__(1024) void chamfer_reduce(
    const float* __restrict__ x, const float* __restrict__ y,
    const unsigned* __restrict__ colKeys, const float* __restrict__ rowRaw,
    float* __restrict__ out) {
    __shared__ float s[1024];
    float acc = 0.0f;
    for (int idx = threadIdx.x; idx < BATCH * NPTS; idx += 1024) {
        const float* xr = x + (size_t)idx * CST;
        const float* yr = y + (size_t)idx * CST;
        float2 xa = *(const float2*)xr;  float xc = xr[2];
        float2 ya = *(const float2*)yr;  float yc = yr[2];
        float rx = fmaf(xc, xc, fmaf(xa.y, xa.y, xa.x * xa.x));
        float ry = fmaf(yc, yc, fmaf(ya.y, ya.y, ya.x * ya.x));
        acc += (decf(colKeys[idx]) + ry) + (rowRaw[idx] + rx);
    }
    s[threadIdx.x] = acc;
    __syncthreads();
    for (int st = 512; st > 0; st >>= 1) {
        if (threadIdx.x < st) s[threadIdx.x] += s[threadIdx.x + st];
        __syncthreads();
    }
    if (threadIdx.x == 0) out[0] = s[0] * (1.0f / (BATCH * NPTS));
}

extern "C" void kernel_launch(void* const* d_in, const int* in_sizes, int n_in,
                              void* d_out, int out_size, void* d_ws, size_t ws_size,
                              hipStream_t stream) {
    const float* x = (const float*)d_in[0];
    const float* y = (const float*)d_in[1];

    unsigned* colKeys = (unsigned*)d_ws;                                   // B*N u32
    float*    rowRaw  = (float*)((char*)d_ws +
                                 (size_t)BATCH * NPTS * sizeof(unsigned)); // B*N f32
    float*    out     = (float*)d_out;

    chamfer_init<<<(BATCH * NPTS + 255) / 256, 256, 0, stream>>>(colKeys);
    chamfer_main<<<BATCH * (NPTS / 128), 256, 0, stream>>>(x, y, colKeys, rowRaw);
    chamfer_reduce<<<1, 1024, 0, stream>>>(x, y, colKeys, rowRaw, out);
}